// RangeDopplerMap_34059090657307
// MI455X (gfx1250) — compile-verified
//
#include <hip/hip_runtime.h>

typedef float v2f __attribute__((ext_vector_type(2)));
typedef float v8f __attribute__((ext_vector_type(8)));

#define PI_F 3.14159265358979323846f

// Modified Bessel I0 via power series: I0(x) = sum_k ((x/2)^k / k!)^2.
// For x <= 25 terms peak ~9e8, sum ~5.8e9 -> safely inside fp32 range.
__device__ __forceinline__ float i0_series(float x) {
  float h = 0.5f * x;
  float term = 1.0f, sum = 1.0f;
  for (int k = 1; k < 64; ++k) {
    float r = h / (float)k;
    term *= r * r;
    sum += term;
  }
  return sum;
}

// Single-workgroup range-Doppler map. Only frame[0] (32 chirps x 64 samples)
// contributes to the reference output, so the whole pipeline collapses to:
//   R(32x64)  = F(32x64) @ M(64x64)      (mean-removal + Hann + rFFT folded)
//   Z(64x32)  = Dbig(64x64) @ Rstk(64x32) (MTI + Kaiser + cFFT + fftshift folded)
// both executed with V_WMMA_F32_16X16X4_F32 (8 waves, one 16x16 C-tile each).
__global__ __launch_bounds__(256) void rdm_kernel(const float* __restrict__ frame,
                                                  float* __restrict__ out) {
  __shared__ float sM[64 * 64];   // range DFT matrix: cols [0,32)=re bins, [32,64)=im bins
  __shared__ float sD[64 * 64];   // doppler matrix [[Dre,-Dim],[Dim,Dre]] w/ MTI+shift folded
  __shared__ float sF[32 * 64];   // frame[0]
  __shared__ float sR[64 * 32];   // [Rre rows(32); Rim rows(32)] x range
  __shared__ float sZ[64 * 32];   // [Zre rows(32); Zim rows(32)] x range
  __shared__ float sRW[64], sDW[32];
  __shared__ float sCre[32], sCsin[32], sScos[32], sSsin[32];
  __shared__ float sNorm[2];

  const int tid    = threadIdx.x;
  const int lane   = tid & 31;
  const int wave   = tid >> 5;
  const int laneLo = lane & 15;
  const int laneHi = lane >> 4;

  const float TH = 2.0f * PI_F / 64.0f;   // range DFT step
  const float OM = 2.0f * PI_F / 32.0f;   // doppler DFT step

  // ---- Phase 0: raw windows; stage frame[0] into LDS ----
  if (tid < 64) {
    sRW[tid] = 0.5f - 0.5f * cosf(2.0f * PI_F * (float)tid / 63.0f);   // Hann, periodic=False
  } else if (tid < 96) {
    int n = tid - 64;
    const float alpha = 15.5f;                                          // (32-1)/2
    float t = ((float)n - alpha) / alpha;
    float arg = fmaxf(1.0f - t * t, 0.0f);
    sDW[n] = i0_series(25.0f * sqrtf(arg)) / i0_series(25.0f);          // Kaiser beta=25
  }
  for (int t = tid; t < 32 * 64; t += 256) sF[t] = frame[t];
  __syncthreads();

  // ---- Phase 1: window sums for unit-sum normalization ----
  if (tid == 0) { float s = 0.f; for (int i = 0; i < 64; ++i) s += sRW[i]; sNorm[0] = s; }
  if (tid == 1) { float s = 0.f; for (int i = 0; i < 32; ++i) s += sDW[i]; sNorm[1] = s; }
  __syncthreads();

  // ---- Phase 2: normalize windows ----
  if (tid < 64)        sRW[tid]      /= sNorm[0];
  else if (tid < 96)   sDW[tid - 64] /= sNorm[1];
  __syncthreads();

  // ---- Phase 3: DC-correction row/column sums (mean removal & MTI folds) ----
  if (tid < 32) {               // Cre[k] = sum_m rw[m] cos(TH k m)
    float s = 0.f;
    for (int m = 0; m < 64; ++m) s += sRW[m] * cosf(TH * (float)(tid * m));
    sCre[tid] = s;
  } else if (tid < 64) {        // Csin[k] = sum_m rw[m] sin(TH k m)
    int k = tid - 32; float s = 0.f;
    for (int m = 0; m < 64; ++m) s += sRW[m] * sinf(TH * (float)(k * m));
    sCsin[k] = s;
  } else if (tid < 96) {        // Scos[d] = sum_c dw[c] cos(OM d c)
    int d = tid - 64; float s = 0.f;
    for (int c = 0; c < 32; ++c) s += sDW[c] * cosf(OM * (float)(d * c));
    sScos[d] = s;
  } else if (tid < 128) {       // Ssin[d] = sum_c dw[c] sin(OM d c)
    int d = tid - 96; float s = 0.f;
    for (int c = 0; c < 32; ++c) s += sDW[c] * sinf(OM * (float)(d * c));
    sSsin[d] = s;
  }
  __syncthreads();

  // ---- Phase 4: build the two folded DFT matrices ----
  for (int t = tid; t < 64 * 64; t += 256) {
    int n = t >> 6, col = t & 63, k = col & 31;
    float ang = TH * (float)(k * n);
    // X[c,k] = sum_n f[c,n] * ( rw[n] e^{-i th k n} - (1/64) sum_m rw[m] e^{-i th k m} )
    sM[t] = (col < 32) ? ( sRW[n] * cosf(ang) - sCre[k]  * (1.0f / 64.0f))
                       : (-sRW[n] * sinf(ang) + sCsin[k] * (1.0f / 64.0f));
  }
  for (int t = tid; t < 64 * 64; t += 256) {
    int row = t >> 6, col = t & 63;
    int d = row & 31, c = col & 31;
    int dd = (d + 16) & 31;                         // fftshift folded into row index
    float ang = OM * (float)(dd * c);
    float dre =  sDW[c] * cosf(ang) - sScos[dd] * (1.0f / 32.0f);  // MTI folded
    float dim = -sDW[c] * sinf(ang) + sSsin[dd] * (1.0f / 32.0f);
    float v;
    if (row < 32) v = (col < 32) ? dre : -dim;      // Zre = Dre@Rre - Dim@Rim
    else          v = (col < 32) ? dim :  dre;      // Zim = Dim@Rre + Dre@Rim
    sD[t] = v;
  }
  __syncthreads();

  // ---- Stage A: R = F @ M via v_wmma_f32_16x16x4_f32 (8 waves, 16 k-steps) ----
  {
    int tileM = wave & 1, tileN = wave >> 1;        // C tiles of 32x64
    v8f acc = {};
    int rowA = (tileM * 16 + laneLo) * 64;
    int colB = tileN * 16 + laneLo;
#pragma unroll
    for (int kb = 0; kb < 64; kb += 4) {
      int ka = kb + 2 * laneHi;                     // A: lanes0-15 hold K=0,1; lanes16-31 K=2,3
      v2f a, b;
      a.x = sF[rowA + ka];
      a.y = sF[rowA + ka + 1];
      b.x = sM[ka * 64 + colB];
      b.y = sM[(ka + 1) * 64 + colB];
      acc = __builtin_amdgcn_wmma_f32_16x16x4_f32(false, a, false, b, (short)0, acc,
                                                  false, false);
    }
#pragma unroll
    for (int r = 0; r < 8; ++r) {                   // D layout: vgpr r -> M=r (+8 hi half)
      int m = tileM * 16 + r + 8 * laneHi;          // chirp
      int n = tileN * 16 + laneLo;                  // [re|im] x range col
      float v = acc[r];
      if (n < 32) sR[m * 32 + n]              = v;  // Rre block
      else        sR[(32 + m) * 32 + (n - 32)] = v; // Rim block
    }
  }
  __syncthreads();

  // ---- Stage B: Z = Dbig @ Rstk via v_wmma_f32_16x16x4_f32 ----
  {
    int tileM = wave >> 1, tileN = wave & 1;        // C tiles of 64x32
    v8f acc = {};
    int rowA = (tileM * 16 + laneLo) * 64;
    int colB = tileN * 16 + laneLo;
#pragma unroll
    for (int kb = 0; kb < 64; kb += 4) {
      int ka = kb + 2 * laneHi;
      v2f a, b;
      a.x = sD[rowA + ka];
      a.y = sD[rowA + ka + 1];
      b.x = sR[ka * 32 + colB];
      b.y = sR[(ka + 1) * 32 + colB];
      acc = __builtin_amdgcn_wmma_f32_16x16x4_f32(false, a, false, b, (short)0, acc,
                                                  false, false);
    }
#pragma unroll
    for (int r = 0; r < 8; ++r) {
      int m = tileM * 16 + r + 8 * laneHi;
      int n = tileN * 16 + laneLo;
      sZ[m * 32 + n] = acc[r];
    }
  }
  __syncthreads();

  // ---- Magnitude, scale, clip, uint8-truncation; flip/transpose layout ----
  // image[i][j] = trunc( clip(1000*mag[d=j][range=31-i], 0, 1) * 255 )
  for (int t = tid; t < 1024; t += 256) {
    int i = t >> 5, j = t & 31;
    int d = j, k = 31 - i;
    float re = sZ[d * 32 + k];
    float im = sZ[(32 + d) * 32 + k];
    float mag = sqrtf(re * re + im * im) * 1000.0f;
    mag = fminf(fmaxf(mag, 0.0f), 1.0f) * 255.0f;
    out[t] = floorf(mag);                           // uint8 cast truncates (non-negative)
  }
}

extern "C" void kernel_launch(void* const* d_in, const int* in_sizes, int n_in,
                              void* d_out, int out_size, void* d_ws, size_t ws_size,
                              hipStream_t stream) {
  (void)in_sizes; (void)n_in; (void)out_size; (void)d_ws; (void)ws_size;
  const float* frame = (const float*)d_in[0];
  float* out = (float*)d_out;
  hipLaunchKernelGGL(rdm_kernel, dim3(1), dim3(256), 0, stream, frame, out);
}